// GraphEncoder_18494129177081
// MI455X (gfx1250) — compile-verified
//
#include <hip/hip_runtime.h>
#include <hip/hip_bf16.h>
#include <math.h>

// Problem constants (from reference): B=4, N=50000, E=800000, D=128
#define BB 4
#define NN 50000
#define EE 800000
#define DD 128
// LDS layout for W: row-pairs interleaved (B-fragment order), padded stride.
// sW[(r>>1)*SW2S + c*2 + (r&1)] = W[r][c]
// 288 mod 64 == 32 -> lane halves (kp vs kp+1) hit disjoint 32-bank windows.
#define SW2S 288

typedef __attribute__((ext_vector_type(2))) float v2f;
typedef __attribute__((ext_vector_type(8))) float v8f;

// ---------------------------------------------------------------------------
// out = x  (initialize running hidden state)
__global__ __launch_bounds__(256) void k_copy4(const float4* __restrict__ src,
                                               float4* __restrict__ dst, long n4) {
  long i = (long)blockIdx.x * blockDim.x + threadIdx.x;
  if (i < n4) dst[i] = src[i];
}

// ---------------------------------------------------------------------------
// degree / normalization (computed once; edges identical for all 3 layers)
__global__ __launch_bounds__(256) void k_zero_deg(unsigned* deg, int n) {
  int i = blockIdx.x * blockDim.x + threadIdx.x;
  if (i < n) deg[i] = 0u;
}

__global__ __launch_bounds__(256) void k_count_deg(const int* __restrict__ eidx,
                                                   unsigned* __restrict__ deg) {
  int e = blockIdx.x * blockDim.x + threadIdx.x;
  int b = blockIdx.y;
  if (e >= EE) return;
  int d = eidx[(size_t)b * 2 * EE + EE + e];  // edge_index[b,1,e] = dst
  atomicAdd(&deg[b * NN + d], 1u);
}

__global__ __launch_bounds__(256) void k_dinv(const unsigned* __restrict__ deg,
                                              float* __restrict__ dinv, int n) {
  int i = blockIdx.x * blockDim.x + threadIdx.x;
  if (i < n) dinv[i] = rsqrtf((float)deg[i] + 1.0f);  // +1 self loop; deg>=1 always
}

// ---------------------------------------------------------------------------
// h_lin = H @ W   (fp32, V_WMMA_F32_16X16X4_F32)
// W (64 KB) staged once per block into LDS, pre-swizzled into B-fragment order
// so each lane's fragment is ONE aligned ds_load_b64 straight into an even
// register pair (no v_mov shuffles). Each wave: 16-row x 128-col strip,
// 32 k-steps x 8 column tiles.
// A 16x4 layout: lane = (K>=2)*16 + M, vgpr = K%2 holds A[M][K]
// B 4x16 layout: lane = (K>=2)*16 + N, vgpr = K%2 holds B[K][N]
// C/D 16x16    : vgpr r, lanes0-15 -> M=r, lanes16-31 -> M=r+8, N=lane&15
__global__ __launch_bounds__(128) void k_gemm_wmma(const float* __restrict__ H,
                                                   const float* __restrict__ W,
                                                   float* __restrict__ O) {
  __shared__ float sW[(DD / 2) * SW2S];  // 64 row-pairs x 288 words = 72 KB

  // cooperative swizzled stage: float4 global reads, 4 scalar LDS writes each
  {
    const float4* Wv = (const float4*)W;
    for (int i = threadIdx.x; i < (DD * DD) / 4; i += blockDim.x) {
      int r = i >> 5;            // row 0..127
      int c4 = (i & 31) * 4;     // starting column of this float4
      float4 v = Wv[(size_t)r * 32 + (i & 31)];
      float* dstp = &sW[(r >> 1) * SW2S + (r & 1)];
      dstp[(c4 + 0) * 2] = v.x;
      dstp[(c4 + 1) * 2] = v.y;
      dstp[(c4 + 2) * 2] = v.z;
      dstp[(c4 + 3) * 2] = v.w;
    }
  }
  __syncthreads();

  const int lane = threadIdx.x & 31;
  const int tile = blockIdx.x * (blockDim.x >> 5) + (threadIdx.x >> 5);  // 16-row tile
  const int batch = blockIdx.y;
  if (tile >= NN / 16) return;  // wave-uniform (after barrier); EXEC all-ones inside

  const float* Hb = H + (size_t)batch * NN * DD;
  float* Ob = O + (size_t)batch * NN * DD;
  const int row0 = tile * 16;

  const int m  = lane & 15;
  const int kq = (lane >> 4) << 1;  // 0 or 2
  const int nI = lane & 15;

  v8f acc[8];
  v8f zero = {};
#pragma unroll
  for (int t = 0; t < 8; ++t) acc[t] = zero;

  const float* arow = Hb + (size_t)(row0 + m) * DD;
#pragma unroll 2
  for (int k0 = 0; k0 < DD; k0 += 4) {
    v2f a = *(const v2f*)(arow + k0 + kq);  // A[m][k0+kq], A[m][k0+kq+1]
    // row-pair index for this lane half: k0/2 (+1 for upper half)
    const float* wr = &sW[((k0 + kq) >> 1) * SW2S + nI * 2];
#pragma unroll
    for (int t = 0; t < 8; ++t) {
      v2f bf = *(const v2f*)(wr + t * 32);  // {W[k0+kq][t*16+nI], W[k0+kq+1][t*16+nI]}
      acc[t] = __builtin_amdgcn_wmma_f32_16x16x4_f32(
          /*neg_a=*/false, a, /*neg_b=*/false, bf,
          /*c_mod=*/(short)0, acc[t], /*reuse_a=*/false, /*reuse_b=*/false);
    }
  }

  const int mh = (lane >> 4) * 8;
#pragma unroll
  for (int t = 0; t < 8; ++t)
#pragma unroll
    for (int r = 0; r < 8; ++r)
      Ob[(size_t)(row0 + mh + r) * DD + t * 16 + nI] = acc[t][r];
}

// ---------------------------------------------------------------------------
// agg[n,:] = h_lin[n,:] * dinv[n]^2   (self-loop term; also zero-replaces ws poison)
__global__ __launch_bounds__(256) void k_agg_init(const float4* __restrict__ hlin,
                                                  const float* __restrict__ dinv,
                                                  float4* __restrict__ agg, long n4) {
  long i = (long)blockIdx.x * blockDim.x + threadIdx.x;
  if (i >= n4) return;
  long node = i >> 5;  // 32 float4 per 128-wide row -> global (b*N+n)
  float w = dinv[node];
  w *= w;
  float4 h = hlin[i];
  agg[i] = make_float4(h.x * w, h.y * w, h.z * w, h.w * w);
}

// ---------------------------------------------------------------------------
// edge scatter: one wave per edge; lane handles 4 consecutive features.
// agg (102 MB) is L2-resident (192 MB L2) -> atomics resolve in L2.
__global__ __launch_bounds__(256) void k_scatter(const float* __restrict__ hlin,
                                                 const int* __restrict__ eidx,
                                                 const float* __restrict__ dinv,
                                                 float* __restrict__ agg) {
  const int lane = threadIdx.x & 31;
  const int e = (int)(((long)blockIdx.x * blockDim.x + threadIdx.x) >> 5);
  const int b = blockIdx.y;
  if (e >= EE) return;
  const int* eb = eidx + (size_t)b * 2 * EE;
  int s = eb[e];
  int d = eb[EE + e];
  const float* db = dinv + (size_t)b * NN;
  float norm = db[s] * db[d];
  const float4 hv = *(const float4*)(hlin + ((size_t)b * NN + s) * DD + lane * 4);
  float* dst = agg + ((size_t)b * NN + d) * DD + lane * 4;
  atomicAdd(dst + 0, hv.x * norm);
  atomicAdd(dst + 1, hv.y * norm);
  atomicAdd(dst + 2, hv.z * norm);
  atomicAdd(dst + 3, hv.w * norm);
}

// ---------------------------------------------------------------------------
// h += act(agg + bias); exact GELU for layers 0,1, identity for layer 2.
__device__ __forceinline__ float gelu_exact(float x) {
  return 0.5f * x * (1.0f + erff(x * 0.70710678118654752f));
}

__global__ __launch_bounds__(256) void k_combine(float4* __restrict__ out,
                                                 const float4* __restrict__ agg,
                                                 const float* __restrict__ bias,
                                                 int applyGelu, long n4) {
  long i = (long)blockIdx.x * blockDim.x + threadIdx.x;
  if (i >= n4) return;
  const float4 bi = *(const float4*)(bias + (int)(i & 31) * 4);
  float4 a = agg[i];
  float4 o = out[i];
  float vx = a.x + bi.x, vy = a.y + bi.y, vz = a.z + bi.z, vw = a.w + bi.w;
  if (applyGelu) {
    vx = gelu_exact(vx); vy = gelu_exact(vy); vz = gelu_exact(vz); vw = gelu_exact(vw);
  }
  out[i] = make_float4(o.x + vx, o.y + vy, o.z + vz, o.w + vw);
}

// ---------------------------------------------------------------------------
extern "C" void kernel_launch(void* const* d_in, const int* in_sizes, int n_in,
                              void* d_out, int out_size, void* d_ws, size_t ws_size,
                              hipStream_t stream) {
  const float* x = (const float*)d_in[0];
  const int* eidx = (const int*)d_in[1];
  const float* Wl[3] = {(const float*)d_in[2], (const float*)d_in[4], (const float*)d_in[6]};
  const float* bl[3] = {(const float*)d_in[3], (const float*)d_in[5], (const float*)d_in[7]};
  float* out = (float*)d_out;

  const size_t nBND = (size_t)BB * NN * DD;            // 25.6M floats
  float* h_lin = (float*)d_ws;                         // 102.4 MB
  float* agg = h_lin + nBND;                           // 102.4 MB
  float* dinv = agg + nBND;                            // 0.8 MB
  unsigned* deg = (unsigned*)(dinv + (size_t)BB * NN); // 0.8 MB

  const long n4 = (long)(nBND / 4);
  const int bn = BB * NN;

  // init hidden state and (once) the symmetric normalization
  k_copy4<<<(unsigned)((n4 + 255) / 256), 256, 0, stream>>>((const float4*)x, (float4*)out, n4);
  k_zero_deg<<<(bn + 255) / 256, 256, 0, stream>>>(deg, bn);
  k_count_deg<<<dim3((EE + 255) / 256, BB), 256, 0, stream>>>(eidx, deg);
  k_dinv<<<(bn + 255) / 256, 256, 0, stream>>>(deg, dinv, bn);

  for (int l = 0; l < 3; ++l) {
    dim3 gemm_grid((NN / 16 + 3) / 4, BB);                 // 4 waves/block (128 thr)
    k_gemm_wmma<<<gemm_grid, 128, 0, stream>>>(out, Wl[l], h_lin);
    k_agg_init<<<(unsigned)((n4 + 255) / 256), 256, 0, stream>>>(
        (const float4*)h_lin, dinv, (float4*)agg, n4);
    k_scatter<<<dim3(EE / 8, BB), 256, 0, stream>>>(h_lin, eidx, dinv, agg);
    k_combine<<<(unsigned)((n4 + 255) / 256), 256, 0, stream>>>(
        (float4*)out, (const float4*)agg, bl[l], l < 2 ? 1 : 0, n4);
  }
}